// BiasedInterpretedFlockingModel_53644141527384
// MI455X (gfx1250) — compile-verified
//
#include <hip/hip_runtime.h>

// ---------------------------------------------------------------------------
// BiasedInterpretedFlockingModel for MI455X (gfx1250)
// Graph scatter/gather: TDM async edge-index staging + L2-resident f32 atomics
// ---------------------------------------------------------------------------

#define TILE 1024          // edges per TDM tile (src+dst rows => 16KB LDS per buffer)

typedef unsigned int        u32;
typedef unsigned long long  u64;
typedef __attribute__((ext_vector_type(4))) unsigned int v4u;
typedef __attribute__((ext_vector_type(4))) int          v4i;
typedef __attribute__((ext_vector_type(8))) int          v8i;

// Low 32 bits of a flat LDS-aperture pointer == workgroup-relative LDS byte
// offset (ISA 10.2).
__device__ __forceinline__ u32 lds_off(const void* p) {
  return (u32)(u64)(uintptr_t)p;
}

__device__ __forceinline__ void fadd(float* p, float v) {
#if defined(__HIP_PLATFORM_AMD__) && defined(__HIP_DEVICE_COMPILE__)
  unsafeAtomicAdd(p, v);           // -> global_atomic_add_f32 (no CAS loop)
#else
  atomicAdd(p, v);
#endif
}

#if __has_builtin(__builtin_amdgcn_tensor_load_to_lds) && __has_builtin(__builtin_amdgcn_s_wait_tensorcnt)
#define USE_TDM 1
// Issue one TDM descriptor: 2D tile (n x 2) of 8-byte elements, rows separated
// by rowStrideElems, packed contiguously into LDS at ldsOff (row0 then row1).
__device__ __forceinline__ void tdm_issue(const long long* gbase, u32 ldsOff,
                                          u32 n, u64 rowStrideElems) {
  u64 ga = (u64)(uintptr_t)gbase;
  v4u g0;
  g0[0] = 1u;                                         // count=1, user D#, no gather
  g0[1] = ldsOff;                                     // lds_addr
  g0[2] = (u32)ga;                                    // global_addr[31:0]
  g0[3] = (u32)((ga >> 32) & 0x01FFFFFFull) | (2u << 30); // global_addr[56:32] | type=2
  v8i g1;
  g1[0] = (int)(3u << 16);                            // wg_mask=0, data_size=3 (8B)
  g1[1] = (int)((n & 0xFFFFu) << 16);                 // tensor_dim0[15:0]
  g1[2] = (int)(((n >> 16) & 0xFFFFu) | (2u << 16));  // tensor_dim0[31:16], tensor_dim1=2
  g1[3] = (int)((n & 0xFFFFu) << 16);                 // tile_dim0 = n
  g1[4] = (int)2;                                     // tile_dim1=2, tile_dim2=0
  g1[5] = (int)(u32)rowStrideElems;                   // tensor_dim0_stride[31:0]
  g1[6] = (int)(u32)((rowStrideElems >> 32) & 0xFFFFull); // dim0_stride[47:32]
  g1[7] = 0;
  v4i z4 = {0, 0, 0, 0};
  v8i z8 = {0, 0, 0, 0, 0, 0, 0, 0};
  // 6-arg form (clang-23 / therock-10.0 headers): groups 0-3 + extra group + cpol
  __builtin_amdgcn_tensor_load_to_lds(g0, g1, z4, z4, z8, 0);
}
#else
#define USE_TDM 0
#endif

// --- per-edge message math + scatter ---------------------------------------
__device__ __forceinline__ void process_edge(int s, int dn,
    const float4* __restrict__ h,
    float* __restrict__ aM0, float* __restrict__ aM1,
    float* __restrict__ aM2, float* __restrict__ aM3,
    float* __restrict__ aC) {
  float4 hj = h[s];    // h_j = h[src]
  float4 hi = h[dn];   // h_i = h[dst]
  float x0  = hi.x - hj.x;
  float x1  = hi.y - hj.y;
  float dv0 = hi.z - hj.z;
  float dv1 = hi.w - hj.w;

  float t  = x0 * 0.07104663f;
  float m0 = (x0 - x1 / (t * t + 1.536996f)) * -0.028956918f;
  float s1 = x0 * -0.021992652f;
  float m1 = (x0 - x1 * (0.8290067f - s1 * s1)) * 0.025425926f;
  float q  = x0 * -0.083299406f;
  float m2 = (x0 - q * q) * -0.024002103f - 0.22298379f;
  float m3 = (x1 + 2.6200492f + x0 * -0.16023761f) * 0.025031794f;

  bool zero = (x0 == 0.0f) & (x1 == 0.0f) & (dv0 == 0.0f) & (dv1 == 0.0f);
  if (zero) { m0 = 0.0f; m1 = 0.0f; m2 = 0.0f; m3 = 0.0f; }

  fadd(&aM0[dn], m0);
  fadd(&aM1[dn], m1);
  fadd(&aM2[dn], m2);
  fadd(&aM3[dn], m3);
  fadd(&aC[dn], 1.0f);   // cnt counts ALL edges (mask does not apply)
}

// --- kernel 1: pack node features + zero accumulators ----------------------
__global__ void init_kernel(const float2* __restrict__ pos,
                            const float2* __restrict__ vel,
                            float4* __restrict__ h,
                            float* __restrict__ acc, int n) {
  int i = blockIdx.x * blockDim.x + threadIdx.x;
  if (i < n) {
    float2 p = pos[i];
    float2 v = vel[i];
    h[i] = make_float4(p.x, p.y, v.x, v.y);
    acc[i]         = 0.0f;
    acc[i + n]     = 0.0f;
    acc[i + 2 * n] = 0.0f;
    acc[i + 3 * n] = 0.0f;
    acc[i + 4 * n] = 0.0f;
  }
}

// --- kernel 2: edge scatter with TDM double-buffered staging ---------------
__global__ void edge_kernel(const long long* __restrict__ eidx,
                            const float4* __restrict__ h,
                            float* __restrict__ acc, int nNodes, int nEdges) {
  float* aM0 = acc;
  float* aM1 = acc + nNodes;
  float* aM2 = acc + 2 * nNodes;
  float* aM3 = acc + 3 * nNodes;
  float* aC  = acc + 4 * nNodes;
#if USE_TDM
  __shared__ long long sbuf[2][2 * TILE];   // [buffer][ src row | dst row ] = 32KB
  const int tiles  = (nEdges + TILE - 1) / TILE;
  const int stride = gridDim.x;
  int tile = blockIdx.x;
  if (tile >= tiles) return;
  int cur = 0;
  if (threadIdx.x < 32) {                    // wave 0 drives the TDM
    int n0 = min(TILE, nEdges - tile * TILE);
    tdm_issue(eidx + (long long)tile * TILE, lds_off(&sbuf[0][0]), (u32)n0, (u64)nEdges);
  }
  while (tile < tiles) {
    int nextTile = tile + stride;
    if (threadIdx.x < 32) {
      if (nextTile < tiles) {                // prefetch next tile into other buffer
        int nn = min(TILE, nEdges - nextTile * TILE);
        tdm_issue(eidx + (long long)nextTile * TILE, lds_off(&sbuf[cur ^ 1][0]),
                  (u32)nn, (u64)nEdges);
        __builtin_amdgcn_s_wait_tensorcnt(1);   // oldest (current tile) complete
      } else {
        __builtin_amdgcn_s_wait_tensorcnt(0);
      }
    }
    __syncthreads();                         // publish LDS to all 8 waves
    const int base = tile * TILE;
    const int n    = min(TILE, nEdges - base);
    for (int e = threadIdx.x; e < n; e += blockDim.x) {
      int s  = (int)sbuf[cur][e];
      int dn = (int)sbuf[cur][n + e];
      process_edge(s, dn, h, aM0, aM1, aM2, aM3, aC);
    }
    __syncthreads();                         // buffer safe to overwrite next iter
    cur ^= 1;
    tile = nextTile;
  }
#else
  for (int e = blockIdx.x * blockDim.x + threadIdx.x; e < nEdges;
       e += gridDim.x * blockDim.x) {
    int s  = (int)eidx[e];
    int dn = (int)eidx[nEdges + e];
    process_edge(s, dn, h, aM0, aM1, aM2, aM3, aC);
  }
#endif
}

// --- kernel 3: node update -------------------------------------------------
__global__ void node_kernel(const float* __restrict__ acc,
                            float* __restrict__ out, int n) {
  int i = blockIdx.x * blockDim.x + threadIdx.x;
  if (i >= n) return;
  float cnt = acc[i + 4 * n];
  float inv = 1.0f / fmaxf(cnt, 1.0f);
  float y0 = acc[i + 2 * n];        // sum m2
  float y1 = acc[i + 3 * n];        // sum m3
  float y2 = acc[i] * inv;          // mean m0
  float y3 = acc[i + n] * inv;      // mean m1

  float t0 = y2 * 0.15994334f;
  float u0 = (y0 - (y3 + t0 * t0) / 1.7044706f - y2) * 0.16596459f;
  float t1 = y2 * -0.089175865f;
  float u1 = (y1 - t1 * t1 * y3 - y2 + y3) * -0.05459863f;
  float u2 = (y3 + y0) * 0.05392959f;
  float u3 = y2 * (12.305774f / (y2 * y2 + 63.129406f));
  float p0 = ((u0 / 0.5268826f + u3 - u2) * -0.18549965f - (u1 + u2)) / 0.7328953f;
  float p1 = u0 * -0.8037861f - u1 + (u3 * 1.2175907f + u2);
  out[2 * i]     = p0;
  out[2 * i + 1] = p1;
}

// ---------------------------------------------------------------------------
extern "C" void kernel_launch(void* const* d_in, const int* in_sizes, int n_in,
                              void* d_out, int out_size, void* d_ws, size_t ws_size,
                              hipStream_t stream) {
  const float2*    pos  = (const float2*)d_in[0];
  const float2*    vel  = (const float2*)d_in[1];
  const long long* eidx = (const long long*)d_in[2];   // int64, shape (2, E)
  const int nNodes = in_sizes[0] / 2;
  const int nEdges = in_sizes[2] / 2;
  float* out = (float*)d_out;

  // workspace: [ h: nNodes * float4 | acc: 5 * nNodes floats ]
  char*   ws  = (char*)d_ws;
  float4* h   = (float4*)ws;
  float*  acc = (float*)(ws + (size_t)nNodes * sizeof(float4));

  const int nb = (nNodes + 255) / 256;
  init_kernel<<<nb, 256, 0, stream>>>(pos, vel, h, acc, nNodes);

  const int tiles  = (nEdges + TILE - 1) / TILE;
  const int eblocks = tiles < 1024 ? tiles : 1024;
  edge_kernel<<<eblocks, 256, 0, stream>>>(eidx, h, acc, nNodes, nEdges);

  node_kernel<<<nb, 256, 0, stream>>>(acc, out, nNodes);
}